// GNNModel_15075335209370
// MI455X (gfx1250) — compile-verified
//
#include <hip/hip_runtime.h>
#include <math.h>

// ---------------------------------------------------------------------------
// GNN (2-layer GAT) for MI455X / gfx1250, wave32.
// GEMMs use native fp32 WMMA (V_WMMA_F32_16X16X4_F32) -> exact fp32 semantics.
// Edge softmax/scatter uses fp32 global atomics; all hot buffers are L2-sized.
// ---------------------------------------------------------------------------

#define NN    100000
#define EE    1600000
#define INDIM 128
#define DD    64
#define HHD   2               // heads
#define GG    64
#define NCLS  4
#define EP    (EE + NN)       // edges + self loops

typedef __attribute__((ext_vector_type(2))) float v2f;
typedef __attribute__((ext_vector_type(8))) float v8f;

__device__ __forceinline__ v8f wmma4f32(v2f a, v2f b, v8f c) {
  // 8 args: (neg_a, A, neg_b, B, c_mod, C, reuse_a, reuse_b)
  return __builtin_amdgcn_wmma_f32_16x16x4_f32(false, a, false, b, (short)0, c,
                                               false, false);
}

__device__ __forceinline__ float leaky02(float v) { return v > 0.f ? v : 0.2f * v; }

// float atomic max via sign-aware int/uint punning (init must be -FLT_MAX)
__device__ __forceinline__ void atomicMaxF(float* a, float v) {
  if (v >= 0.f) atomicMax((int*)a, __float_as_int(v));
  else          atomicMin((unsigned int*)a, __float_as_uint(v));
}

__device__ __forceinline__ void edge_sd(const int* __restrict__ ei, int e,
                                        int& s, int& d) {
  if (e < EE) { s = ei[e]; d = ei[EE + e]; }
  else        { s = e - EE; d = s; }        // self loop
}

// ---------------------------------------------------------------------------
// Projection: h_in[N,64] = [x | emb[ids]] @ proj_W[192,64] + proj_b
// One wave per 16x16 output tile; K split at the x/emb boundary (aligned).
// ---------------------------------------------------------------------------
__global__ __launch_bounds__(32)
void proj_gemm(const float* __restrict__ x, const int* __restrict__ ids,
               const float* __restrict__ emb, const float* __restrict__ W,
               const float* __restrict__ b, float* __restrict__ out) {
  const int lane = threadIdx.x;
  const int row0 = blockIdx.x * 16;
  const int col0 = blockIdx.y * 16;
  const int m    = lane & 15;               // A row / B-C col within tile
  const int kg   = (lane >> 4) * 2;         // K sub-offset per lane half
  const float* xrow = x + (size_t)(row0 + m) * INDIM;
  const float* erow = emb + (size_t)ids[row0 + m] * DD;
  v8f c = {0.f, 0.f, 0.f, 0.f, 0.f, 0.f, 0.f, 0.f};
  for (int k = 0; k < INDIM; k += 4) {      // x part of the concat
    int ka = k + kg;
    v2f a = {xrow[ka], xrow[ka + 1]};
    v2f bb = {W[(size_t)ka * DD + col0 + m], W[(size_t)(ka + 1) * DD + col0 + m]};
    c = wmma4f32(a, bb, c);
  }
  for (int k = 0; k < DD; k += 4) {         // emb part of the concat
    int ka = k + kg;
    int kw = INDIM + ka;
    v2f a = {erow[ka], erow[ka + 1]};
    v2f bb = {W[(size_t)kw * DD + col0 + m], W[(size_t)(kw + 1) * DD + col0 + m]};
    c = wmma4f32(a, bb, c);
  }
  const int half = (lane >> 4) * 8;
  const float bias = b[col0 + m];
#pragma unroll
  for (int v = 0; v < 8; ++v)
    out[(size_t)(row0 + half + v) * DD + col0 + m] = c[v] + bias;
}

// ---------------------------------------------------------------------------
// Layer GEMM: hh[N,128] = h_in[N,64] @ W[64,128]   (per-head features)
// ---------------------------------------------------------------------------
__global__ __launch_bounds__(32)
void layer_gemm(const float* __restrict__ hin, const float* __restrict__ W,
                float* __restrict__ hh) {
  const int lane = threadIdx.x;
  const int row0 = blockIdx.x * 16;
  const int col0 = blockIdx.y * 16;
  const int m    = lane & 15;
  const int kg   = (lane >> 4) * 2;
  const float* arow = hin + (size_t)(row0 + m) * DD;
  v8f c = {0.f, 0.f, 0.f, 0.f, 0.f, 0.f, 0.f, 0.f};
  for (int k = 0; k < DD; k += 4) {
    int ka = k + kg;
    v2f a = {arow[ka], arow[ka + 1]};
    v2f bb = {W[(size_t)ka * (HHD * DD) + col0 + m],
              W[(size_t)(ka + 1) * (HHD * DD) + col0 + m]};
    c = wmma4f32(a, bb, c);
  }
  const int half = (lane >> 4) * 8;
#pragma unroll
  for (int v = 0; v < 8; ++v)
    hh[(size_t)(row0 + half + v) * (HHD * DD) + col0 + m] = c[v];
}

// ---------------------------------------------------------------------------
// Attention logits per (node, head): a_s = <h, att_src>, a_d = <h, att_dst>
// ---------------------------------------------------------------------------
__global__ void att_logits(const float* __restrict__ hh,
                           const float* __restrict__ att_src,
                           const float* __restrict__ att_dst,
                           float* __restrict__ a_s, float* __restrict__ a_d) {
  int t = blockIdx.x * blockDim.x + threadIdx.x;
  if (t >= NN * HHD) return;
  int n = t >> 1, h = t & 1;
  const float4* hv = (const float4*)(hh + (size_t)n * (HHD * DD) + h * DD);
  const float4* as = (const float4*)(att_src + h * DD);
  const float4* ad = (const float4*)(att_dst + h * DD);
  float s0 = 0.f, s1 = 0.f;
#pragma unroll
  for (int i = 0; i < DD / 4; ++i) {
    float4 q = hv[i], u = as[i], w = ad[i];
    s0 += q.x * u.x + q.y * u.y + q.z * u.z + q.w * u.w;
    s1 += q.x * w.x + q.y * w.y + q.z * w.z + q.w * w.w;
  }
  a_s[t] = s0;
  a_d[t] = s1;
}

__global__ void fill_kernel(float* __restrict__ p, float v, long long n) {
  long long i = (long long)blockIdx.x * blockDim.x + threadIdx.x;
  long long stride = (long long)gridDim.x * blockDim.x;
  for (; i < n; i += stride) p[i] = v;
}

// ---------------------------------------------------------------------------
// Edge pass A: segment max of leaky_relu logits over dst
// ---------------------------------------------------------------------------
__global__ void edge_max(const int* __restrict__ ei, const float* __restrict__ a_s,
                         const float* __restrict__ a_d, float* __restrict__ mmax) {
  int t = blockIdx.x * blockDim.x + threadIdx.x;
  if (t >= EP * HHD) return;
  int e = t >> 1, h = t & 1;
  int s, d; edge_sd(ei, e, s, d);
  float v = leaky02(a_s[s * HHD + h] + a_d[d * HHD + h]);
  atomicMaxF(&mmax[d * HHD + h], v);
}

// ---------------------------------------------------------------------------
// Edge pass B: denom[dst] += exp(v - max[dst])
// ---------------------------------------------------------------------------
__global__ void edge_denom(const int* __restrict__ ei, const float* __restrict__ a_s,
                           const float* __restrict__ a_d, const float* __restrict__ mmax,
                           float* __restrict__ den) {
  int t = blockIdx.x * blockDim.x + threadIdx.x;
  if (t >= EP * HHD) return;
  int e = t >> 1, h = t & 1;
  int s, d; edge_sd(ei, e, s, d);
  int di = d * HHD + h;
  float v = leaky02(a_s[s * HHD + h] + a_d[di]);
  atomicAdd(&den[di], __expf(v - mmax[di]));
}

// ---------------------------------------------------------------------------
// Edge pass C: acc[dst,h,:] += alpha * hh[src,h,:]   (alpha recomputed)
// hh and acc (51 MB each) are both L2-resident on MI455X (192 MB L2).
// ---------------------------------------------------------------------------
__global__ void edge_scatter(const int* __restrict__ ei, const float* __restrict__ a_s,
                             const float* __restrict__ a_d, const float* __restrict__ mmax,
                             const float* __restrict__ den, const float* __restrict__ hh,
                             float* __restrict__ acc) {
  int t = blockIdx.x * blockDim.x + threadIdx.x;
  if (t >= EP * HHD) return;
  int e = t >> 1, h = t & 1;
  int s, d; edge_sd(ei, e, s, d);
  int di = d * HHD + h;
  float v = leaky02(a_s[s * HHD + h] + a_d[di]);
  float w = __expf(v - mmax[di]) / den[di];
  const float4* src = (const float4*)(hh + (size_t)s * (HHD * DD) + h * DD);
  float* dst = acc + (size_t)d * (HHD * DD) + h * DD;
#pragma unroll
  for (int i = 0; i < DD / 4; ++i) {
    float4 q = src[i];
    atomicAdd(dst + 4 * i + 0, w * q.x);
    atomicAdd(dst + 4 * i + 1, w * q.y);
    atomicAdd(dst + 4 * i + 2, w * q.z);
    atomicAdd(dst + 4 * i + 3, w * q.w);
  }
}

// head mean + bias + relu -> layer output
__global__ void finalize_layer(const float* __restrict__ acc, const float* __restrict__ b,
                               float* __restrict__ hout) {
  int t = blockIdx.x * blockDim.x + threadIdx.x;
  if (t >= NN * DD) return;
  int n = t >> 6, k = t & 63;
  float v = 0.5f * (acc[(size_t)n * (HHD * DD) + k] +
                    acc[(size_t)n * (HHD * DD) + DD + k]) + b[k];
  hout[t] = fmaxf(v, 0.f);
}

// global max pool per graph
__global__ void pool_max(const float* __restrict__ hin, const int* __restrict__ batch,
                         float* __restrict__ pooled) {
  int t = blockIdx.x * blockDim.x + threadIdx.x;
  if (t >= NN * DD) return;
  int n = t >> 6, k = t & 63;
  atomicMaxF(&pooled[batch[n] * DD + k], hin[t]);
}

// [G,64] @ [64,4] + b  (tiny)
__global__ void classifier(const float* __restrict__ pooled, const float* __restrict__ W,
                           const float* __restrict__ b, float* __restrict__ out) {
  int t = blockIdx.x * blockDim.x + threadIdx.x;
  if (t >= GG * NCLS) return;
  int g = t >> 2, c = t & 3;
  float s = b[c];
  for (int k = 0; k < DD; ++k) s += pooled[g * DD + k] * W[k * NCLS + c];
  out[t] = s;
}

// ---------------------------------------------------------------------------
extern "C" void kernel_launch(void* const* d_in, const int* in_sizes, int n_in,
                              void* d_out, int out_size, void* d_ws, size_t ws_size,
                              hipStream_t stream) {
  const float* x        = (const float*)d_in[0];
  const int*   ids      = (const int*)d_in[1];
  const int*   ei       = (const int*)d_in[2];
  const int*   batch    = (const int*)d_in[3];
  const float* emb      = (const float*)d_in[4];
  const float* proj_W   = (const float*)d_in[5];
  const float* proj_b   = (const float*)d_in[6];
  const float* W1       = (const float*)d_in[7];
  const float* as1      = (const float*)d_in[8];
  const float* ad1      = (const float*)d_in[9];
  const float* b1       = (const float*)d_in[10];
  const float* W2       = (const float*)d_in[11];
  const float* as2      = (const float*)d_in[12];
  const float* ad2      = (const float*)d_in[13];
  const float* b2       = (const float*)d_in[14];
  const float* cls_W    = (const float*)d_in[15];
  const float* cls_b    = (const float*)d_in[16];

  // workspace: N*(64+128+128+8) + 64*64 floats  (~131 MB)
  float* ws     = (float*)d_ws;
  float* h_in   = ws;                               // [N,64]
  float* hh     = h_in + (size_t)NN * DD;           // [N,H*D]
  float* acc    = hh + (size_t)NN * HHD * DD;       // [N,H*D]
  float* a_s    = acc + (size_t)NN * HHD * DD;      // [N,H]
  float* a_d    = a_s + (size_t)NN * HHD;           // [N,H]
  float* mmax   = a_d + (size_t)NN * HHD;           // [N,H]
  float* den    = mmax + (size_t)NN * HHD;          // [N,H]
  float* pooled = den + (size_t)NN * HHD;           // [G,64]

  const int eb = (EP * HHD + 255) / 256;            // edge-pass blocks
  const int nb = (NN * DD + 255) / 256;             // node*dim blocks

  // projection (fp32 WMMA)
  proj_gemm<<<dim3(NN / 16, DD / 16), 32, 0, stream>>>(x, ids, emb, proj_W, proj_b, h_in);

  const float* Ws[2]  = {W1, W2};
  const float* ass[2] = {as1, as2};
  const float* ads[2] = {ad1, ad2};
  const float* bs[2]  = {b1, b2};
  for (int layer = 0; layer < 2; ++layer) {
    layer_gemm<<<dim3(NN / 16, (HHD * DD) / 16), 32, 0, stream>>>(h_in, Ws[layer], hh);
    att_logits<<<(NN * HHD + 255) / 256, 256, 0, stream>>>(hh, ass[layer], ads[layer], a_s, a_d);
    fill_kernel<<<512, 256, 0, stream>>>(mmax, -3.402823466e38f, (long long)NN * HHD);
    fill_kernel<<<512, 256, 0, stream>>>(den, 0.f, (long long)NN * HHD);
    fill_kernel<<<4096, 256, 0, stream>>>(acc, 0.f, (long long)NN * HHD * DD);
    edge_max<<<eb, 256, 0, stream>>>(ei, a_s, a_d, mmax);
    edge_denom<<<eb, 256, 0, stream>>>(ei, a_s, a_d, mmax, den);
    edge_scatter<<<eb, 256, 0, stream>>>(ei, a_s, a_d, mmax, den, hh, acc);
    finalize_layer<<<nb, 256, 0, stream>>>(acc, bs[layer], h_in);
  }

  fill_kernel<<<16, 256, 0, stream>>>(pooled, -3.402823466e38f, (long long)GG * DD);
  pool_max<<<nb, 256, 0, stream>>>(h_in, batch, pooled);
  classifier<<<1, 256, 0, stream>>>(pooled, cls_W, cls_b, (float*)d_out);
}